// TriplaneDensityField_83202106458409
// MI455X (gfx1250) — compile-verified
//
#include <hip/hip_runtime.h>
#include <hip/hip_bf16.h>

// Triplane density field for MI455X (gfx1250).
//
// Roofline: ~0.9 GFLOP -> compute-trivial; gather-bound. All three grids
// (96 MB) fit in the 192 MB L2, so gathers are L2 hits after warmup. Keys:
//  1. Transpose grids [32][512][512] -> [512*512][32]: all 32 channels of a
//     texel become one 128B cacheline.
//  2. Fast kernel: 4 channels per lane (float4), 8 lanes per point, 4 points
//     per wave iteration -> each bilinear corner is ONE global_load_b128
//     covering 4 points; 3 vmem instructions per point total.
//  3. Point coords staged into LDS via CDNA5 async-to-LDS + s_wait_asynccnt.
//  4. Channel mean: 3 in-register adds + 3-step 8-lane shfl_xor butterfly,
//     results gathered so the final store is one coalesced 128B write/wave.

#ifndef __has_builtin
#define __has_builtin(x) 0
#endif

typedef __attribute__((address_space(1))) int GlobalI32;
typedef __attribute__((address_space(3))) int LdsI32;

#if __has_builtin(__builtin_amdgcn_global_load_async_to_lds_b32)
#define HAVE_ASYNC_LDS 1
__device__ __forceinline__ void async_cp4(const float* g, float* l) {
    __builtin_amdgcn_global_load_async_to_lds_b32(
        (GlobalI32*)(g), (LdsI32*)(l), 0, 0);
}
#elif __has_builtin(__builtin_amdgcn_load_to_lds)
#define HAVE_ASYNC_LDS 1
__device__ __forceinline__ void async_cp4(const float* g, float* l) {
    __builtin_amdgcn_load_to_lds((GlobalI32*)(g), (LdsI32*)(l), 4, 0, 0);
}
#else
#define HAVE_ASYNC_LDS 0
#endif

__device__ __forceinline__ void wait_async_lds() {
#if __has_builtin(__builtin_amdgcn_s_wait_asynccnt)
    __builtin_amdgcn_s_wait_asynccnt(0);
#else
    asm volatile("s_wait_asynccnt 0" ::: "memory");
#endif
}

#define RES 512
#define RANK 32
#define PLANE_ELEMS ((long)RANK * RES * RES)   // 8388608
#define N_PTS (8192L * 128L)                   // 1048576
#define WAVES_PER_BLOCK 8
#define PTS_PER_WAVE 32

// -------------------------------------------------------------------------
// Transpose [32][512][512] -> [512*512][32]. Reads lane-coalesced; each lane
// writes its own 128B line as 8x float4 (L2 write-combines).
// -------------------------------------------------------------------------
__global__ __launch_bounds__(256) void transpose_kernel(
    const float* __restrict__ in, float* __restrict__ out) {
    const long PL = (long)RES * RES;
    long pos = (long)blockIdx.x * 256 + threadIdx.x;   // 0..262143
    float4* o = (float4*)(out + pos * RANK);
#pragma unroll
    for (int cb = 0; cb < RANK; cb += 4) {
        float4 t;
        t.x = in[(long)(cb + 0) * PL + pos];
        t.y = in[(long)(cb + 1) * PL + pos];
        t.z = in[(long)(cb + 2) * PL + pos];
        t.w = in[(long)(cb + 3) * PL + pos];
        o[cb >> 2] = t;
    }
}

// -------------------------------------------------------------------------
// Texel coordinate helper (matches reference align_corners=True + clipping).
// a -> W index, b -> H index. Emits x0,x1,y0,y1 row offsets and weights.
// -------------------------------------------------------------------------
struct Coord2D {
    int i00, i01, i10, i11;   // texel indices y*512+x
    float w00, w01, w10, w11;
};

__device__ __forceinline__ Coord2D coord2d(float a, float b) {
    float u = (a + 1.0f) * (0.5f * (RES - 1));
    float v = (b + 1.0f) * (0.5f * (RES - 1));
    float uf = floorf(u), vf = floorf(v);
    float wx = u - uf, wy = v - vf;
    int x0 = (int)uf;  x0 = x0 < 0 ? 0 : (x0 > RES - 1 ? RES - 1 : x0);
    int y0 = (int)vf;  y0 = y0 < 0 ? 0 : (y0 > RES - 1 ? RES - 1 : y0);
    int x1 = x0 + 1 > RES - 1 ? RES - 1 : x0 + 1;
    int y1 = y0 + 1 > RES - 1 ? RES - 1 : y0 + 1;
    int r0 = y0 << 9, r1 = y1 << 9;
    Coord2D c;
    c.i00 = r0 + x0;  c.i01 = r0 + x1;  c.i10 = r1 + x0;  c.i11 = r1 + x1;
    c.w00 = (1.0f - wx) * (1.0f - wy);
    c.w01 = wx * (1.0f - wy);
    c.w10 = (1.0f - wx) * wy;
    c.w11 = wx * wy;
    return c;
}

// float4 bilinear from transposed layout: texel t's channels at g[t*32..].
// sub selects this lane's channel quad (0..7). One b128 load per corner.
__device__ __forceinline__ float4 bilin4(const float* __restrict__ g,
                                         const Coord2D& c, int sub) {
    const float4* p00 = (const float4*)(g + (long)c.i00 * RANK) + sub;
    const float4* p01 = (const float4*)(g + (long)c.i01 * RANK) + sub;
    const float4* p10 = (const float4*)(g + (long)c.i10 * RANK) + sub;
    const float4* p11 = (const float4*)(g + (long)c.i11 * RANK) + sub;
    float4 f00 = *p00, f01 = *p01, f10 = *p10, f11 = *p11;
    float4 r;
    r.x = fmaf(f00.x, c.w00, fmaf(f01.x, c.w01, fmaf(f10.x, c.w10, f11.x * c.w11)));
    r.y = fmaf(f00.y, c.w00, fmaf(f01.y, c.w01, fmaf(f10.y, c.w10, f11.y * c.w11)));
    r.z = fmaf(f00.z, c.w00, fmaf(f01.z, c.w01, fmaf(f10.z, c.w10, f11.z * c.w11)));
    r.w = fmaf(f00.w, c.w00, fmaf(f01.w, c.w01, fmaf(f10.w, c.w10, f11.w * c.w11)));
    return r;
}

// -------------------------------------------------------------------------
// Fast kernel (transposed layout): 8 lanes per point, 4 points per wave
// iteration, 32 points per wave.
// -------------------------------------------------------------------------
__global__ __launch_bounds__(256) void triplane_fast_kernel(
    const float* __restrict__ pts,
    const float* __restrict__ gxy, const float* __restrict__ gxz,
    const float* __restrict__ gyz,
    const float* __restrict__ aabb, float* __restrict__ out) {
    __shared__ float spts[WAVES_PER_BLOCK][PTS_PER_WAVE * 3];  // 3 KB

    const int lane = threadIdx.x & 31;
    const int wv = threadIdx.x >> 5;
    const int g4 = lane >> 3;      // which of 4 points in this iteration
    const int sub = lane & 7;      // channel quad: channels 4*sub..4*sub+3
    const long pBase = ((long)blockIdx.x * WAVES_PER_BLOCK + wv) * PTS_PER_WAVE;

    // Stage this wave's 96 point-coord floats into LDS via async copies.
    const float* src = pts + pBase * 3;
#if HAVE_ASYNC_LDS
#pragma unroll
    for (int i = 0; i < 3; ++i) {
        int idx = lane + i * 32;
        async_cp4(src + idx, &spts[wv][idx]);
    }
    wait_async_lds();
#else
#pragma unroll
    for (int i = 0; i < 3; ++i) {
        int idx = lane + i * 32;
        spts[wv][idx] = src[idx];
    }
#endif

    float lo0 = aabb[0], lo1 = aabb[1], lo2 = aabb[2];
    float s0 = 2.0f / (aabb[3] - lo0);
    float s1 = 2.0f / (aabb[4] - lo1);
    float s2 = 2.0f / (aabb[5] - lo2);

    float myDensity = 0.0f;

#pragma unroll 2
    for (int it = 0; it < PTS_PER_WAVE / 4; ++it) {
        // This lane's point within the iteration.
        int kp = (it << 2) + g4;
        float px = (spts[wv][kp * 3 + 0] - lo0) * s0 - 1.0f;
        float py = (spts[wv][kp * 3 + 1] - lo1) * s1 - 1.0f;
        float pz = (spts[wv][kp * 3 + 2] - lo2) * s2 - 1.0f;

        // xy: (x->W, y->H); xz: (x->W, z->H); yz: (y->W, z->H)
        Coord2D cxy = coord2d(px, py);
        Coord2D cxz = coord2d(px, pz);
        Coord2D cyz = coord2d(py, pz);

        float4 a = bilin4(gxy, cxy, sub);
        float4 b = bilin4(gxz, cxz, sub);
        float4 c = bilin4(gyz, cyz, sub);

        // 4-channel partial product-sum in registers.
        float s = a.x * b.x * c.x + a.y * b.y * c.y
                + a.z * b.z * c.z + a.w * b.w * c.w;

        // 8-lane butterfly: every lane in the group holds the 32-chan sum.
        s += __shfl_xor(s, 1, 32);
        s += __shfl_xor(s, 2, 32);
        s += __shfl_xor(s, 4, 32);

        // Route point (it*4+j)'s sum to output lane it*4+j.
        float gsum = __shfl(s, (lane & 3) << 3, 32);
        if ((lane >> 2) == it) myDensity = gsum;
    }

    // One coalesced 128B store per wave.
    out[pBase + lane] = expf(myDensity * (1.0f / RANK));
}

// -------------------------------------------------------------------------
// Generic fallback (original [C][H][W] layout): lane == channel, scalar.
// -------------------------------------------------------------------------
__device__ __forceinline__ float bilin1(const float* __restrict__ g,
                                        const Coord2D& c, long laneOff,
                                        int posStride) {
    float f00 = g[(long)c.i00 * posStride + laneOff];
    float f01 = g[(long)c.i01 * posStride + laneOff];
    float f10 = g[(long)c.i10 * posStride + laneOff];
    float f11 = g[(long)c.i11 * posStride + laneOff];
    return fmaf(f00, c.w00, fmaf(f01, c.w01, fmaf(f10, c.w10, f11 * c.w11)));
}

__global__ __launch_bounds__(256) void triplane_generic_kernel(
    const float* __restrict__ pts,
    const float* __restrict__ gxy, const float* __restrict__ gxz,
    const float* __restrict__ gyz,
    const float* __restrict__ aabb, float* __restrict__ out,
    int chanStride, int posStride) {
    __shared__ float spts[WAVES_PER_BLOCK][PTS_PER_WAVE * 3];

    const int lane = threadIdx.x & 31;
    const int wv = threadIdx.x >> 5;
    const long pBase = ((long)blockIdx.x * WAVES_PER_BLOCK + wv) * PTS_PER_WAVE;
    const float* src = pts + pBase * 3;
#pragma unroll
    for (int i = 0; i < 3; ++i) {
        int idx = lane + i * 32;
        spts[wv][idx] = src[idx];
    }

    float lo0 = aabb[0], lo1 = aabb[1], lo2 = aabb[2];
    float s0 = 2.0f / (aabb[3] - lo0);
    float s1 = 2.0f / (aabb[4] - lo1);
    float s2 = 2.0f / (aabb[5] - lo2);

    const long laneOff = (long)lane * chanStride;
    float myDensity = 0.0f;

    for (int k = 0; k < PTS_PER_WAVE; ++k) {
        float px = (spts[wv][k * 3 + 0] - lo0) * s0 - 1.0f;
        float py = (spts[wv][k * 3 + 1] - lo1) * s1 - 1.0f;
        float pz = (spts[wv][k * 3 + 2] - lo2) * s2 - 1.0f;
        Coord2D cxy = coord2d(px, py);
        Coord2D cxz = coord2d(px, pz);
        Coord2D cyz = coord2d(py, pz);
        float v = bilin1(gxy, cxy, laneOff, posStride)
                * bilin1(gxz, cxz, laneOff, posStride)
                * bilin1(gyz, cyz, laneOff, posStride);
#pragma unroll
        for (int off = 16; off; off >>= 1) v += __shfl_xor(v, off, 32);
        if (lane == k) myDensity = v;
    }
    out[pBase + lane] = expf(myDensity * (1.0f / RANK));
}

// -------------------------------------------------------------------------
extern "C" void kernel_launch(void* const* d_in, const int* in_sizes, int n_in,
                              void* d_out, int out_size, void* d_ws, size_t ws_size,
                              hipStream_t stream) {
    const float* pts  = (const float*)d_in[0];
    const float* gxy  = (const float*)d_in[1];
    const float* gxz  = (const float*)d_in[2];
    const float* gyz  = (const float*)d_in[3];
    const float* aabb = (const float*)d_in[4];
    float* out = (float*)d_out;

    const size_t needBytes = (size_t)3 * PLANE_ELEMS * sizeof(float);  // ~96 MB
    const int mainBlocks = (int)(N_PTS / (WAVES_PER_BLOCK * PTS_PER_WAVE));  // 4096

    if (ws_size >= needBytes) {
        float* t = (float*)d_ws;
        const int tBlocks = (int)((RES * RES) / 256);  // 1024
        transpose_kernel<<<tBlocks, 256, 0, stream>>>(gxy, t);
        transpose_kernel<<<tBlocks, 256, 0, stream>>>(gxz, t + PLANE_ELEMS);
        transpose_kernel<<<tBlocks, 256, 0, stream>>>(gyz, t + 2 * PLANE_ELEMS);
        triplane_fast_kernel<<<mainBlocks, 256, 0, stream>>>(
            pts, t, t + PLANE_ELEMS, t + 2 * PLANE_ELEMS, aabb, out);
    } else {
        triplane_generic_kernel<<<mainBlocks, 256, 0, stream>>>(
            pts, gxy, gxz, gyz, aabb, out,
            /*chanStride=*/RES * RES, /*posStride=*/1);
    }
}